// Task_prob_with_pre_85272280695307
// MI455X (gfx1250) — compile-verified
//
#include <hip/hip_runtime.h>

typedef float v2f __attribute__((ext_vector_type(2)));
typedef float v8f __attribute__((ext_vector_type(8)));
typedef int   vi4 __attribute__((vector_size(16)));
typedef int   vi2 __attribute__((vector_size(8)));

#define Dd     512          // feature dim
#define KCc    2048         // total centers (K*C)
#define OUTK   256          // output groups per row (KCc / 8)
#define BM     64
#define BN     128
#define BK     32
#define NSTAGE (Dd / BK)    // 16
#define ASL    36           // A LDS row stride (floats), padded, 16B-aligned rows
#define BSL    144          // B LDS stride per k-pair row (float2 units), bank-split halves

#if defined(__AMDGCN__) && __has_builtin(__builtin_amdgcn_global_load_async_to_lds_b128) && \
    __has_builtin(__builtin_amdgcn_global_load_async_to_lds_b64)
#define USE_ASYNC 1
#else
#define USE_ASYNC 0
#endif

#if USE_ASYNC
__device__ __forceinline__ void async_b128(const float* g, float* l) {
    __builtin_amdgcn_global_load_async_to_lds_b128(
        (__attribute__((address_space(1))) vi4*)g,
        (__attribute__((address_space(3))) vi4*)l, 0, 0);
}
__device__ __forceinline__ void async_b64(const float* g, float2* l) {
    __builtin_amdgcn_global_load_async_to_lds_b64(
        (__attribute__((address_space(1))) vi2*)g,
        (__attribute__((address_space(3))) vi2*)l, 0, 0);
}
__device__ __forceinline__ void wait_async0() {
#if __has_builtin(__builtin_amdgcn_s_wait_asynccnt)
    __builtin_amdgcn_s_wait_asynccnt(0);
#else
    asm volatile("s_wait_asynccnt 0" ::: "memory");
#endif
}
#endif

// ---------------- row squared-norm kernel: one wave per row ----------------
__global__ __launch_bounds__(256)
void rownorm_kernel(const float* __restrict__ A, float* __restrict__ out, int nrows) {
    int row  = blockIdx.x * 8 + (threadIdx.x >> 5);
    int lane = threadIdx.x & 31;
    if (row >= nrows) return;
    const float* p = A + (size_t)row * Dd;
    float s = 0.f;
    for (int i = lane; i < Dd; i += 32) { float v = p[i]; s = fmaf(v, v, s); }
    for (int m = 16; m >= 1; m >>= 1) s += __shfl_xor(s, m, 32);
    if (lane == 0) out[row] = s;
}

// ---------------- fused distance-GEMM + gaussian + group-reduce ----------------
__global__ __launch_bounds__(256)
void gauss_kernel(const float* __restrict__ Xa,   // x        [Brows, Dd]
                  const float* __restrict__ Xb,   // pre_out  [Brows, Dd]
                  const float* __restrict__ Cf,   // centers  [KCc, Dd]
                  const float* __restrict__ x2b,  // [2, Brows]
                  const float* __restrict__ c2,   // [KCc]
                  float* __restrict__ outp,       // [2, Brows, OUTK]
                  int Brows)
{
    __shared__ __align__(16) float  As[2][BM * ASL];
    __shared__ __align__(16) float2 Bs[2][(BK / 2) * BSL];

    const int tid   = threadIdx.x;
    const int lane  = tid & 31;
    const int wave  = tid >> 5;
    const int half  = lane >> 4;      // 0: lanes 0-15, 1: lanes 16-31
    const int lm    = lane & 15;
    const int waveM = wave & 3;       // 0..3  (16 rows each)
    const int waveN = wave >> 2;      // 0..1  (64 cols each)

    const int z = blockIdx.z;
    const float* __restrict__ X  = z ? Xb : Xa;
    const float* __restrict__ x2 = x2b + (size_t)z * Brows;
    float* __restrict__ O        = outp + (size_t)z * Brows * OUTK;

    const int m0 = blockIdx.x * BM;
    const int n0 = blockIdx.y * BN;

    // global-load coordinates: 16B chunks per thread
    const int arow = tid >> 3;        // 0..31
    const int acq  = (tid & 7) * 4;   // k-offset within stage (floats)

    const float* Xg = X  + (size_t)(m0 + arow) * Dd + acq;

    v8f acc[4] = {};
    const int amoff = (waveM * 16 + lm) * ASL + 2 * half;
    const int nb    = waveN * 64;

    auto compute_stage = [&](int buf) {
        const float*  Ab = As[buf];
        const float2* Bb = Bs[buf];
#pragma unroll
        for (int dk = 0; dk < BK / 4; ++dk) {
            // A fragment: lanes 0-15 hold {K0,K1}, lanes 16-31 hold {K2,K3}
            v2f a = *(const v2f*)(Ab + amoff + dk * 4);
#pragma unroll
            for (int tn = 0; tn < 4; ++tn) {
                float2 t = Bb[(dk * 2 + half) * BSL + nb + tn * 16 + lm];
                v2f b; b.x = t.x; b.y = t.y;
                acc[tn] = __builtin_amdgcn_wmma_f32_16x16x4_f32(
                    false, a, false, b, (short)0, acc[tn], false, false);
            }
        }
    };

#if USE_ASYNC
    // ---- async global->LDS path: transpose-scatter done by per-lane LDS addressing ----
    const float* Cg2 = Cf + (size_t)(n0 + wave * 16 + half) * Dd + lm * 2;

    auto issue_stage = [&](int ks, int buf) {
        // A tile: two 16B chunks per thread into padded rows
        async_b128(Xg + ks, &As[buf][arow * ASL + acq]);
        async_b128(Xg + (size_t)32 * Dd + ks, &As[buf][(arow + 32) * ASL + acq]);
        // B tile: 8 float2 per thread; lane's k-pair lands in pair-row lm at column n
        float2* db = &Bs[buf][lm * BSL];
#pragma unroll
        for (int j = 0; j < 8; ++j) {
            const int n = wave * 16 + j * 2 + half;
            async_b64(Cg2 + (size_t)(j * 2) * Dd + ks, db + n);
        }
    };

    issue_stage(0, 0);
    wait_async0();
    __syncthreads();

#pragma unroll 1
    for (int s = 0; s < NSTAGE; ++s) {
        if (s + 1 < NSTAGE) issue_stage((s + 1) * BK, (s + 1) & 1);
        compute_stage(s & 1);
        if (s + 1 < NSTAGE) {
            wait_async0();
            __syncthreads();
        }
    }
#else
    // ---- fallback: manual load -> register -> LDS store path ----
    const float* Cg = Cf + (size_t)(n0 + arow) * Dd + acq;
    float4 pa0, pa1, pb0, pb1, pb2, pb3;

    auto load_stage = [&](int ks) {
        pa0 = *(const float4*)(Xg + ks);
        pa1 = *(const float4*)(Xg + (size_t)32 * Dd + ks);
        pb0 = *(const float4*)(Cg + ks);
        pb1 = *(const float4*)(Cg + (size_t)32 * Dd + ks);
        pb2 = *(const float4*)(Cg + (size_t)64 * Dd + ks);
        pb3 = *(const float4*)(Cg + (size_t)96 * Dd + ks);
    };

    auto store_stage = [&](int buf) {
        *(float4*)&As[buf][arow * ASL + acq]        = pa0;
        *(float4*)&As[buf][(arow + 32) * ASL + acq] = pa1;
        float2* B = Bs[buf];
        const int p = acq >> 1;
        B[(p + 0) * BSL + arow     ] = make_float2(pb0.x, pb0.y);
        B[(p + 1) * BSL + arow     ] = make_float2(pb0.z, pb0.w);
        B[(p + 0) * BSL + arow + 32] = make_float2(pb1.x, pb1.y);
        B[(p + 1) * BSL + arow + 32] = make_float2(pb1.z, pb1.w);
        B[(p + 0) * BSL + arow + 64] = make_float2(pb2.x, pb2.y);
        B[(p + 1) * BSL + arow + 64] = make_float2(pb2.z, pb2.w);
        B[(p + 0) * BSL + arow + 96] = make_float2(pb3.x, pb3.y);
        B[(p + 1) * BSL + arow + 96] = make_float2(pb3.z, pb3.w);
    };

    load_stage(0);
    store_stage(0);
    __syncthreads();

#pragma unroll 1
    for (int s = 0; s < NSTAGE; ++s) {
        if (s + 1 < NSTAGE) load_stage((s + 1) * BK);
        compute_stage(s & 1);
        if (s + 1 < NSTAGE) {
            store_stage((s + 1) & 1);
            __syncthreads();
        }
    }
#endif

    // ---------- epilogue: d2 -> exp -> 8-lane group sum/max -> P ----------
    const int rowbase = m0 + waveM * 16 + 8 * half;   // D layout: M = r + 8*half
    float x2v[8];
#pragma unroll
    for (int r = 0; r < 8; ++r) x2v[r] = x2[rowbase + r];

#pragma unroll
    for (int tn = 0; tn < 4; ++tn) {
        const int col = n0 + nb + tn * 16 + lm;
        const float c2v = c2[col];
        const int kg = (n0 + nb + tn * 16 + (lm & 8)) >> 3;
#pragma unroll
        for (int r = 0; r < 8; ++r) {
            float s  = acc[tn][r];
            float d2 = x2v[r] + c2v - 2.0f * s;
            float g  = __expf(-0.5f * d2);
            float sg = g, mg = g;
            // reduce across the 8-lane column group (lanes differing in bits 0..2)
            sg += __shfl_xor(sg, 1, 32);
            mg  = fmaxf(mg, __shfl_xor(mg, 1, 32));
            sg += __shfl_xor(sg, 2, 32);
            mg  = fmaxf(mg, __shfl_xor(mg, 2, 32));
            sg += __shfl_xor(sg, 4, 32);
            mg  = fmaxf(mg, __shfl_xor(mg, 4, 32));
            if ((lane & 7) == 0) {
                float P = sg / (sg + 8.0f - 8.0f * mg);
                O[(size_t)(rowbase + r) * OUTK + kg] = P;
            }
        }
    }
}

extern "C" void kernel_launch(void* const* d_in, const int* in_sizes, int n_in,
                              void* d_out, int out_size, void* d_ws, size_t ws_size,
                              hipStream_t stream) {
    const float* x       = (const float*)d_in[0];
    const float* pre     = (const float*)d_in[1];
    const float* centers = (const float*)d_in[2];  // [K, C, Dd] flat = [KCc, Dd]
    float* out = (float*)d_out;
    float* ws  = (float*)d_ws;

    const int Brows = in_sizes[0] / Dd;   // 8192
    const int nc    = in_sizes[2] / Dd;   // 2048

    float* x2 = ws;                 // [Brows]
    float* p2 = ws + Brows;         // [Brows]
    float* c2 = ws + 2 * Brows;     // [nc]

    rownorm_kernel<<<(Brows + 7) / 8, 256, 0, stream>>>(x, x2, Brows);
    rownorm_kernel<<<(Brows + 7) / 8, 256, 0, stream>>>(pre, p2, Brows);
    rownorm_kernel<<<(nc + 7) / 8, 256, 0, stream>>>(centers, c2, nc);

    dim3 grid(Brows / BM, KCc / BN, 2);
    gauss_kernel<<<grid, 256, 0, stream>>>(x, pre, centers, x2, c2, out, Brows);
}